// Cell_82729660056407
// MI455X (gfx1250) — compile-verified
//
#include <hip/hip_runtime.h>
#include <hip/hip_bf16.h>

// ---------------------------------------------------------------------------
// DARTS cell (C=1) on MI455X / gfx1250, wave32.
// Memory-bound plan: recompute-don't-materialize; 3 streaming phases per step
// (BN dependency chain), states kept fp32 in ws (L2-resident: <48MB of 192MB).
// CDNA5 paths: v_wmma_f32_16x16x4_f32 in the wave-reduction epilogue,
// global_prefetch_b8, wave32 shuffles, LDS-staged stencils.
// ---------------------------------------------------------------------------

#define TPB   256
#define ITEMS 16
#define TILE  (TPB * ITEMS)          // 4096 elements per block
#define HALO  4                      // max stencil reach (dil5 / sep5)
#define LROW  32768
#define NROWS 64
#define NELEM (LROW * NROWS)         // 2,097,152
#define TILES_PER_ROW (LROW / TILE)  // 8
#define NBLK  (NELEM / TILE)         // 512
#define EPS   1e-5f

typedef __attribute__((ext_vector_type(2))) float v2f;
typedef __attribute__((ext_vector_type(8))) float v8f;

// ---- wave32 sum via WMMA ones-trick + xor-shuffle fold ---------------------
// A = ones(16x4), B vgpr0 = B vgpr1 = per-lane partial p.
// Every D element is a column sum = 2*(p_n + p_{n+16})  (column n <-> lanes
// {n, n+16}); 4 xor-shuffles fold the 16 columns; *0.5 gives the wave total.
__device__ __forceinline__ float wave_sum32(float v) {
  v2f a; a[0] = 1.0f; a[1] = 1.0f;
  v2f b; b[0] = v;    b[1] = v;
  v8f c = {0.f, 0.f, 0.f, 0.f, 0.f, 0.f, 0.f, 0.f};
  c = __builtin_amdgcn_wmma_f32_16x16x4_f32(false, a, false, b, (short)0, c,
                                            false, false);
  float s = c[0];
  s += __shfl_xor(s, 1, 32);
  s += __shfl_xor(s, 2, 32);
  s += __shfl_xor(s, 4, 32);
  s += __shfl_xor(s, 8, 32);
  return 0.5f * s;
}

template <int NS>
__device__ __forceinline__ void block_reduce_atomic(float* __restrict__ dst,
                                                    const float (&vals)[NS],
                                                    float* red) {
  const int lane = threadIdx.x & 31;
  const int wv   = threadIdx.x >> 5;
#pragma unroll
  for (int k = 0; k < NS; ++k) {
    float s = wave_sum32(vals[k]);   // uniform control flow: EXEC all ones
    if (lane == 0) red[wv * NS + k] = s;
  }
  __syncthreads();
  if (threadIdx.x < NS) {
    float t = 0.f;
#pragma unroll
    for (int w = 0; w < TPB / 32; ++w) t += red[w * NS + threadIdx.x];
    atomicAdd(dst + threadIdx.x, t);  // one global_atomic_add_f32 per stat/block
  }
  __syncthreads();
}

// ---- tile + halo into LDS; element e of thread t lives at e*TPB+t ----------
__device__ __forceinline__ void load_tile(const float* __restrict__ h, int row,
                                          int t0, float* tile) {
  const float* rowp = h + (size_t)row * LROW;
#pragma unroll
  for (int e = 0; e < ITEMS; ++e) {
    int li = e * TPB + threadIdx.x;
    tile[HALO + li] = rowp[t0 + li];        // tiles divide the row exactly
  }
  const int tid = threadIdx.x;
  if (tid < HALO) {
    int gl = t0 - HALO + tid;
    tile[tid] = (gl >= 0) ? rowp[gl] : 0.0f;               // zero pad (conv)
    int gr = t0 + TILE + tid;
    tile[HALO + TILE + tid] = (gr < LROW) ? rowp[gr] : 0.0f;
  }
}

// BN(c*u) = alpha*(u - mean),  alpha = c * rsqrt(c^2*var + eps)
__device__ __forceinline__ void bn_coef(const float* __restrict__ st, float pw,
                                        float& alpha, float& mean) {
  const float invn = 1.0f / (float)NELEM;
  float m  = st[0] * invn;
  float e2 = st[1] * invn;
  float var = fmaxf(e2 - m * m, 0.0f);
  alpha = pw * rsqrtf(pw * pw * var + EPS);
  mean  = m;
}

// ---------------------------------------------------------------------------
__global__ void zero_stats_kernel(float* p, int n) {
  if ((int)threadIdx.x < n) p[threadIdx.x] = 0.0f;
}

// ---- Phase A: stats of conv1 outputs (sep3, sep5, dil3, dil5) --------------
__global__ __launch_bounds__(TPB) void phaseA_kernel(
    const float* __restrict__ x, const float* __restrict__ sbuf,
    const float* __restrict__ s3dw1, const float* __restrict__ s5dw1,
    const float* __restrict__ d3dw, const float* __restrict__ d5dw,
    float* __restrict__ statsA, int step, int eoff) {
  __shared__ float tile[TILE + 2 * HALO];
  __shared__ float red[(TPB / 32) * 8];
  const int j   = blockIdx.y;
  const int eid = eoff + j;
  const float* h = (j == 0) ? x : (sbuf + (size_t)(j - 1) * NELEM);
  const int row = blockIdx.x / TILES_PER_ROW;
  const int t0  = (blockIdx.x % TILES_PER_ROW) * TILE;
  load_tile(h, row, t0, tile);
  __syncthreads();

  const float* k3  = s3dw1 + step * 3;
  const float* k5  = s5dw1 + step * 5;
  const float* kd3 = d3dw + step * 3;
  const float* kd5 = d5dw + step * 5;

  float acc[8] = {0, 0, 0, 0, 0, 0, 0, 0};
#pragma unroll
  for (int e = 0; e < ITEMS; ++e) {
    int li = e * TPB + threadIdx.x;
    float rm4 = fmaxf(tile[HALO + li - 4], 0.f);
    float rm2 = fmaxf(tile[HALO + li - 2], 0.f);
    float rm1 = fmaxf(tile[HALO + li - 1], 0.f);
    float rc  = fmaxf(tile[HALO + li    ], 0.f);
    float rp1 = fmaxf(tile[HALO + li + 1], 0.f);
    float rp2 = fmaxf(tile[HALO + li + 2], 0.f);
    float rp4 = fmaxf(tile[HALO + li + 4], 0.f);
    float u3  = k3[0] * rm1 + k3[1] * rc + k3[2] * rp1;
    float u5  = k5[0] * rm2 + k5[1] * rm1 + k5[2] * rc + k5[3] * rp1 + k5[4] * rp2;
    float ud3 = kd3[0] * rm2 + kd3[1] * rc + kd3[2] * rp2;
    float ud5 = kd5[0] * rm4 + kd5[1] * rm2 + kd5[2] * rc + kd5[3] * rp2 + kd5[4] * rp4;
    acc[0] += u3;  acc[1] += u3 * u3;
    acc[2] += u5;  acc[3] += u5 * u5;
    acc[4] += ud3; acc[5] += ud3 * ud3;
    acc[6] += ud5; acc[7] += ud5 * ud5;
  }
  block_reduce_atomic<8>(statsA + eid * 8, acc, red);
}

// ---- Phase B: stats of conv2 outputs (sep3, sep5) --------------------------
__global__ __launch_bounds__(TPB) void phaseB_kernel(
    const float* __restrict__ x, const float* __restrict__ sbuf,
    const float* __restrict__ s3dw1, const float* __restrict__ s3pw1,
    const float* __restrict__ s3dw2, const float* __restrict__ s5dw1,
    const float* __restrict__ s5pw1, const float* __restrict__ s5dw2,
    const float* __restrict__ statsA, float* __restrict__ statsB, int step,
    int eoff) {
  __shared__ float tile[TILE + 2 * HALO];
  __shared__ float stage[TILE + 2 * HALO];
  __shared__ float red[(TPB / 32) * 4];
  const int j   = blockIdx.y;
  const int eid = eoff + j;
  const float* h = (j == 0) ? x : (sbuf + (size_t)(j - 1) * NELEM);
  const int row = blockIdx.x / TILES_PER_ROW;
  const int t0  = (blockIdx.x % TILES_PER_ROW) * TILE;
  load_tile(h, row, t0, tile);
  __syncthreads();

  const float* k3a = s3dw1 + step * 3;
  const float* k3b = s3dw2 + step * 3;
  const float* k5a = s5dw1 + step * 5;
  const float* k5b = s5dw2 + step * 5;
  float al3, m3, al5, m5;
  bn_coef(statsA + eid * 8 + 0, s3pw1[step], al3, m3);
  bn_coef(statsA + eid * 8 + 2, s5pw1[step], al5, m5);

  float acc[4] = {0, 0, 0, 0};
  // stage = relu(BN1(conv1)) for sep3 over [-1, TILE]
  for (int idx = threadIdx.x; idx < TILE + 2; idx += TPB) {
    int i = idx - 1;
    float rm1 = fmaxf(tile[HALO + i - 1], 0.f);
    float rc  = fmaxf(tile[HALO + i    ], 0.f);
    float rp1 = fmaxf(tile[HALO + i + 1], 0.f);
    float u = k3a[0] * rm1 + k3a[1] * rc + k3a[2] * rp1;
    stage[HALO + i] = fmaxf(al3 * (u - m3), 0.f);
  }
  __syncthreads();
#pragma unroll
  for (int e = 0; e < ITEMS; ++e) {
    int li = e * TPB + threadIdx.x;
    float u = k3b[0] * stage[HALO + li - 1] + k3b[1] * stage[HALO + li] +
              k3b[2] * stage[HALO + li + 1];
    acc[0] += u; acc[1] += u * u;
  }
  __syncthreads();
  // stage = relu(BN1(conv1)) for sep5 over [-2, TILE+1]
  for (int idx = threadIdx.x; idx < TILE + 4; idx += TPB) {
    int i = idx - 2;
    float rm2 = fmaxf(tile[HALO + i - 2], 0.f);
    float rm1 = fmaxf(tile[HALO + i - 1], 0.f);
    float rc  = fmaxf(tile[HALO + i    ], 0.f);
    float rp1 = fmaxf(tile[HALO + i + 1], 0.f);
    float rp2 = fmaxf(tile[HALO + i + 2], 0.f);
    float u = k5a[0] * rm2 + k5a[1] * rm1 + k5a[2] * rc + k5a[3] * rp1 + k5a[4] * rp2;
    stage[HALO + i] = fmaxf(al5 * (u - m5), 0.f);
  }
  __syncthreads();
#pragma unroll
  for (int e = 0; e < ITEMS; ++e) {
    int li = e * TPB + threadIdx.x;
    float u = k5b[0] * stage[HALO + li - 2] + k5b[1] * stage[HALO + li - 1] +
              k5b[2] * stage[HALO + li] + k5b[3] * stage[HALO + li + 1] +
              k5b[4] * stage[HALO + li + 2];
    acc[2] += u; acc[3] += u * u;
  }
  block_reduce_atomic<4>(statsB + eid * 4, acc, red);
}

// ---- Phase C: combine all primitives over all edges of this step -----------
__global__ __launch_bounds__(TPB) void phaseC_kernel(
    const float* __restrict__ x, float* __restrict__ sbuf,
    const float* __restrict__ weights, const float* __restrict__ s3dw1,
    const float* __restrict__ s3pw1, const float* __restrict__ s3dw2,
    const float* __restrict__ s3pw2, const float* __restrict__ s5dw1,
    const float* __restrict__ s5pw1, const float* __restrict__ s5dw2,
    const float* __restrict__ s5pw2, const float* __restrict__ d3dw,
    const float* __restrict__ d3pw, const float* __restrict__ d5dw,
    const float* __restrict__ d5pw, const float* __restrict__ statsA,
    const float* __restrict__ statsB, int step, int eoff) {
  __shared__ float tile[TILE + 2 * HALO];
  __shared__ float stage[TILE + 2 * HALO];
  const int row = blockIdx.x / TILES_PER_ROW;
  const int t0  = (blockIdx.x % TILES_PER_ROW) * TILE;

  const float* k3a = s3dw1 + step * 3;
  const float* k3b = s3dw2 + step * 3;
  const float* k5a = s5dw1 + step * 5;
  const float* k5b = s5dw2 + step * 5;
  const float* kd3 = d3dw + step * 3;
  const float* kd5 = d5dw + step * 5;

  float acc[ITEMS];
#pragma unroll
  for (int e = 0; e < ITEMS; ++e) acc[e] = 0.f;

  for (int j = 0; j <= step; ++j) {
    const int eid = eoff + j;
    const float* h = (j == 0) ? x : (sbuf + (size_t)(j - 1) * NELEM);
    const float* wrow = weights + eid * 8;
    float w_max = wrow[1], w_avg = wrow[2], w_skip = wrow[3];
    float w_s3 = wrow[4], w_s5 = wrow[5], w_d3 = wrow[6], w_d5 = wrow[7];

    float al3, m3, al5, m5, ad3, md3, ad5, md5, b3, mb3, b5, mb5;
    bn_coef(statsA + eid * 8 + 0, s3pw1[step], al3, m3);
    bn_coef(statsA + eid * 8 + 2, s5pw1[step], al5, m5);
    bn_coef(statsA + eid * 8 + 4, d3pw[step], ad3, md3);
    bn_coef(statsA + eid * 8 + 6, d5pw[step], ad5, md5);
    bn_coef(statsB + eid * 4 + 0, s3pw2[step], b3, mb3);
    bn_coef(statsB + eid * 4 + 2, s5pw2[step], b5, mb5);

    load_tile(h, row, t0, tile);
    // prefetch next edge's tile into cache (global_prefetch_b8)
    if (j < step && threadIdx.x < 8) {
      const float* hn = sbuf + (size_t)j * NELEM;   // next edge input (j+1>=1)
      __builtin_prefetch(hn + (size_t)row * LROW + t0 + threadIdx.x * 512, 0, 3);
    }
    __syncthreads();

    // direct terms: pools, skip, dilated convs
#pragma unroll
    for (int e = 0; e < ITEMS; ++e) {
      int li = e * TPB + threadIdx.x;
      int t  = t0 + li;
      float vm4 = tile[HALO + li - 4], vm2 = tile[HALO + li - 2];
      float vm1 = tile[HALO + li - 1], vc  = tile[HALO + li];
      float vp1 = tile[HALO + li + 1], vp2 = tile[HALO + li + 2];
      float vp4 = tile[HALO + li + 4];
      float mx;
      if (t == 0)              mx = fmaxf(vc, vp1);
      else if (t == LROW - 1)  mx = fmaxf(vm1, vc);
      else                     mx = fmaxf(fmaxf(vm1, vc), vp1);
      float inv = (t == 0 || t == LROW - 1) ? 0.5f : (1.0f / 3.0f);
      float av  = (vm1 + vc + vp1) * inv;       // zero-halo sum is exact
      float rm4 = fmaxf(vm4, 0.f), rm2 = fmaxf(vm2, 0.f), rc = fmaxf(vc, 0.f);
      float rp2 = fmaxf(vp2, 0.f), rp4 = fmaxf(vp4, 0.f);
      float ud3 = kd3[0] * rm2 + kd3[1] * rc + kd3[2] * rp2;
      float ud5 = kd5[0] * rm4 + kd5[1] * rm2 + kd5[2] * rc + kd5[3] * rp2 + kd5[4] * rp4;
      acc[e] += w_max * mx + w_avg * av + w_skip * vc +
                w_d3 * (ad3 * (ud3 - md3)) + w_d5 * (ad5 * (ud5 - md5));
    }

    // sep3: stage relu(BN1(conv1)) then conv2 -> BN2
    for (int idx = threadIdx.x; idx < TILE + 2; idx += TPB) {
      int i = idx - 1;
      float rm1 = fmaxf(tile[HALO + i - 1], 0.f);
      float rc  = fmaxf(tile[HALO + i    ], 0.f);
      float rp1 = fmaxf(tile[HALO + i + 1], 0.f);
      float u = k3a[0] * rm1 + k3a[1] * rc + k3a[2] * rp1;
      stage[HALO + i] = fmaxf(al3 * (u - m3), 0.f);
    }
    __syncthreads();
#pragma unroll
    for (int e = 0; e < ITEMS; ++e) {
      int li = e * TPB + threadIdx.x;
      float u = k3b[0] * stage[HALO + li - 1] + k3b[1] * stage[HALO + li] +
                k3b[2] * stage[HALO + li + 1];
      acc[e] += w_s3 * (b3 * (u - mb3));
    }
    __syncthreads();

    // sep5
    for (int idx = threadIdx.x; idx < TILE + 4; idx += TPB) {
      int i = idx - 2;
      float rm2 = fmaxf(tile[HALO + i - 2], 0.f);
      float rm1 = fmaxf(tile[HALO + i - 1], 0.f);
      float rc  = fmaxf(tile[HALO + i    ], 0.f);
      float rp1 = fmaxf(tile[HALO + i + 1], 0.f);
      float rp2 = fmaxf(tile[HALO + i + 2], 0.f);
      float u = k5a[0] * rm2 + k5a[1] * rm1 + k5a[2] * rc + k5a[3] * rp1 + k5a[4] * rp2;
      stage[HALO + i] = fmaxf(al5 * (u - m5), 0.f);
    }
    __syncthreads();
#pragma unroll
    for (int e = 0; e < ITEMS; ++e) {
      int li = e * TPB + threadIdx.x;
      float u = k5b[0] * stage[HALO + li - 2] + k5b[1] * stage[HALO + li - 1] +
                k5b[2] * stage[HALO + li] + k5b[3] * stage[HALO + li + 1] +
                k5b[4] * stage[HALO + li + 2];
      acc[e] += w_s5 * (b5 * (u - mb5));
    }
    __syncthreads();   // before next edge overwrites tile/stage
  }

  float* sOut = sbuf + (size_t)step * NELEM + (size_t)row * LROW + t0;
#pragma unroll
  for (int e = 0; e < ITEMS; ++e) sOut[e * TPB + threadIdx.x] = acc[e];
}

// ---- out = mean of the 4 states ------------------------------------------
__global__ __launch_bounds__(TPB) void finalize_kernel(
    const float* __restrict__ sbuf, float* __restrict__ out) {
  size_t base = (size_t)blockIdx.x * TILE;
#pragma unroll
  for (int e = 0; e < ITEMS; ++e) {
    size_t idx = base + e * TPB + threadIdx.x;
    out[idx] = 0.25f * (sbuf[idx] + sbuf[idx + (size_t)NELEM] +
                        sbuf[idx + 2 * (size_t)NELEM] +
                        sbuf[idx + 3 * (size_t)NELEM]);
  }
}

// ---------------------------------------------------------------------------
extern "C" void kernel_launch(void* const* d_in, const int* in_sizes, int n_in,
                              void* d_out, int out_size, void* d_ws,
                              size_t ws_size, hipStream_t stream) {
  (void)in_sizes; (void)n_in; (void)out_size; (void)ws_size;
  const float* x     = (const float*)d_in[0];
  const float* wts   = (const float*)d_in[1];
  const float* s3dw1 = (const float*)d_in[2];
  const float* s3pw1 = (const float*)d_in[3];
  const float* s3dw2 = (const float*)d_in[4];
  const float* s3pw2 = (const float*)d_in[5];
  const float* s5dw1 = (const float*)d_in[6];
  const float* s5pw1 = (const float*)d_in[7];
  const float* s5dw2 = (const float*)d_in[8];
  const float* s5pw2 = (const float*)d_in[9];
  const float* d3dw  = (const float*)d_in[10];
  const float* d3pw  = (const float*)d_in[11];
  const float* d5dw  = (const float*)d_in[12];
  const float* d5pw  = (const float*)d_in[13];

  float* S      = (float*)d_ws;
  float* sbuf   = S;                          // 4 states * NELEM
  float* statsA = S + (size_t)4 * NELEM;      // 10 edges * 8
  float* statsB = statsA + 80;                // 10 edges * 4

  zero_stats_kernel<<<1, 128, 0, stream>>>(statsA, 120);

  static const int eoff[4] = {0, 1, 3, 6};
  dim3 blk(TPB);
  for (int i = 0; i < 4; ++i) {
    phaseA_kernel<<<dim3(NBLK, i + 1), blk, 0, stream>>>(
        x, sbuf, s3dw1, s5dw1, d3dw, d5dw, statsA, i, eoff[i]);
    phaseB_kernel<<<dim3(NBLK, i + 1), blk, 0, stream>>>(
        x, sbuf, s3dw1, s3pw1, s3dw2, s5dw1, s5pw1, s5dw2, statsA, statsB, i,
        eoff[i]);
    phaseC_kernel<<<dim3(NBLK), blk, 0, stream>>>(
        x, sbuf, wts, s3dw1, s3pw1, s3dw2, s3pw2, s5dw1, s5pw1, s5dw2, s5pw2,
        d3dw, d3pw, d5dw, d5pw, statsA, statsB, i, eoff[i]);
  }
  finalize_kernel<<<dim3(NBLK), blk, 0, stream>>>(sbuf, (float*)d_out);
}